// VQSign_45475113730159
// MI455X (gfx1250) — compile-verified
//
#include <hip/hip_runtime.h>

// ---------------------------------------------------------------------------
// VQ nearest-codebook quantization for gfx1250 (MI455X, wave32, WMMA).
//   z:        [32*1024, 1024] f32   (d_in[0])
//   codebook: [256, 1024] f32       (d_in[1])
//   out:      quantized [32768*1024] f32  ++  indices [32768] (as f32 tail)
// Core: cross = z @ cb^T via V_WMMA_F32_16X16X4_F32, argmin(c_sq - 2*cross).
// CDNA5 paths: v_wmma_f32_16x16x4_f32 compute, double-buffered
// global_load_async_to_lds_b128 staging (ASYNCcnt), ds_load_b64 fragments.
// ---------------------------------------------------------------------------

typedef float v2f __attribute__((ext_vector_type(2)));
typedef float v8f __attribute__((ext_vector_type(8)));

#define DIMS     1024
#define CODES    256
#define NROWS    32768
#define KC       64            // D-chunk staged in LDS per iteration
#define NCHUNK   (DIMS / KC)
#define LDSTR    (KC + 4)      // 68 floats: stride ≡ 4 (mod 64) -> conflict-free
                               // b64 fragment loads across all 32 lanes
#define ROWS_WG  128           // 8 waves * 16 rows

// --------------------------- kernel 0: ||c_k||^2 ---------------------------
__global__ __launch_bounds__(256)
void vq_csq_kernel(const float* __restrict__ cb, float* __restrict__ csq) {
    const int k   = blockIdx.x;
    const int tid = threadIdx.x;
    const float* row = cb + (size_t)k * DIMS;
    float s = 0.f;
    for (int d = tid; d < DIMS; d += 256) { float v = row[d]; s += v * v; }
    #pragma unroll
    for (int off = 16; off > 0; off >>= 1) s += __shfl_xor(s, off, 32);
    __shared__ float partial[8];
    if ((tid & 31) == 0) partial[tid >> 5] = s;
    __syncthreads();
    if (tid == 0) {
        float t = 0.f;
        #pragma unroll
        for (int w = 0; w < 8; ++w) t += partial[w];
        csq[k] = t;
    }
}

// ----------------- kernel 1: WMMA GEMM + per-row argmin --------------------
__global__ __launch_bounds__(256)
void vq_argmin_kernel(const float* __restrict__ z,
                      const float* __restrict__ cb,
                      const float* __restrict__ csq,
                      int* __restrict__ out_idx) {
    // Double-buffered tiles: 2 * (34.0 KB + 68.0 KB) ~= 204 KB of the 320 KB WGP LDS.
    __shared__ float zs [2][ROWS_WG * LDSTR];
    __shared__ float cbs[2][CODES  * LDSTR];

    const int tid      = threadIdx.x;
    const int lane     = tid & 31;
    const int wave     = tid >> 5;           // 8 waves, 16 rows each
    const int half     = lane >> 4;          // ISA lane-split (K upper half)
    const int l16      = lane & 15;
    const int row_base = blockIdx.x * ROWS_WG;

    const float* zblk = z + (size_t)row_base * DIMS;

    // Issue this thread's async global->LDS copies for one D-chunk.
    // GVS addressing: saddr = uniform base (SGPR pair), vaddr = 32-bit byte off.
    // 8 b128 ops for z (128 rows x 16 float4) + 16 for codebook (256 x 16).
    auto issue_chunk = [&](int kc0, int bufi) {
        #pragma unroll
        for (int i = 0; i < 8; ++i) {
            int e = tid + i * 256;
            int r = e >> 4;                 // 16 float4 per row
            int c = (e & 15) << 2;
            unsigned dst  = (unsigned)(unsigned long long)&zs[bufi][r * LDSTR + c];
            unsigned voff = (unsigned)(((size_t)r * DIMS + kc0 + c) * sizeof(float));
            asm volatile("global_load_async_to_lds_b128 %0, %1, %2"
                         :: "v"(dst), "v"(voff), "s"(zblk) : "memory");
        }
        #pragma unroll
        for (int i = 0; i < 16; ++i) {
            int e = tid + i * 256;
            int r = e >> 4;
            int c = (e & 15) << 2;
            unsigned dst  = (unsigned)(unsigned long long)&cbs[bufi][r * LDSTR + c];
            unsigned voff = (unsigned)(((size_t)r * DIMS + kc0 + c) * sizeof(float));
            asm volatile("global_load_async_to_lds_b128 %0, %1, %2"
                         :: "v"(dst), "v"(voff), "s"(cb) : "memory");
        }
    };

    // 16 accumulator tiles of 16x16 f32 -> this wave's 16 rows x 256 codes.
    v8f acc[16] = {};

    issue_chunk(0, 0);                       // prologue: chunk 0 -> buffer 0

    for (int ch = 0; ch < NCHUNK; ++ch) {
        // Only chunk ch's async ops are outstanding here; drain + barrier =>
        // chunk ch visible in LDS to all waves, and every wave has finished
        // reading buffer (ch-1)&1 == (ch+1)&1, so refilling it is race-free.
        asm volatile("s_wait_asynccnt 0x0" ::: "memory");
        __syncthreads();
        if (ch + 1 < NCHUNK)
            issue_chunk((ch + 1) * KC, (ch + 1) & 1);  // overlaps compute below

        // A-frag (16x4 f32): lane l16 = row; lanes 0-15 hold K=k..k+1,
        // lanes 16-31 hold K=k+2..k+3  -> one ds_load_b64 per step.
        // B-frag (4x16 f32): lane l16 = code-within-tile, same K split.
        const float* za_base = &zs [ch & 1][(wave * 16 + l16) * LDSTR + 2 * half];
        const float* cb_base = &cbs[ch & 1][l16 * LDSTR + 2 * half];

        for (int k = 0; k < KC; k += 4) {
            v2f a = *(const v2f*)(za_base + k);
            #pragma unroll
            for (int t = 0; t < 16; ++t) {
                v2f b = *(const v2f*)(cb_base + t * 16 * LDSTR + k);
                acc[t] = __builtin_amdgcn_wmma_f32_16x16x4_f32(
                    /*neg_a=*/false, a, /*neg_b=*/false, b,
                    /*c_mod=*/(short)0, acc[t],
                    /*reuse_a=*/false, /*reuse_b=*/false);
            }
        }
    }

    // score(row, code) = ||c||^2 - 2 * cross   (z^2 term is argmin-invariant)
    float cs[16];
    #pragma unroll
    for (int t = 0; t < 16; ++t) cs[t] = csq[t * 16 + l16];

    // C/D layout: VGPR v holds M=v (lanes 0-15) and M=v+8 (lanes 16-31), N=l16.
    #pragma unroll
    for (int v = 0; v < 8; ++v) {
        float best = 3.4e38f;
        int   bidx = 0x7fffffff;
        #pragma unroll
        for (int t = 0; t < 16; ++t) {
            float s = cs[t] - 2.0f * acc[t][v];
            if (s < best) { best = s; bidx = t * 16 + l16; }
        }
        // min-reduce across the 16 lanes of this half; ties -> smaller index
        // (matches jnp.argmin first-occurrence semantics).
        #pragma unroll
        for (int off = 1; off < 16; off <<= 1) {
            float ob = __shfl_xor(best, off, 32);
            int   oi = __shfl_xor(bidx, off, 32);
            if (ob < best || (ob == best && oi < bidx)) { best = ob; bidx = oi; }
        }
        if (l16 == 0)
            out_idx[row_base + wave * 16 + half * 8 + v] = bidx;
    }
}

// ------------------- kernel 2: gather + emit outputs -----------------------
__global__ __launch_bounds__(256)
void vq_gather_kernel(const float* __restrict__ cb,
                      const int* __restrict__ idx,
                      float* __restrict__ out_q,
                      float* __restrict__ out_i) {
    const int row  = blockIdx.x;
    const int code = idx[row];                       // scalar (uniform) load
    const float4* src = (const float4*)(cb + (size_t)code * DIMS);
    float4*       dst = (float4*)(out_q + (size_t)row * DIMS);
    dst[threadIdx.x] = src[threadIdx.x];             // 256 lanes x 16B = 4 KB row
    if (threadIdx.x == 0) out_i[row] = (float)code;  // tuple tail (promoted f32)
}

// ---------------------------------------------------------------------------
extern "C" void kernel_launch(void* const* d_in, const int* in_sizes, int n_in,
                              void* d_out, int out_size, void* d_ws, size_t ws_size,
                              hipStream_t stream) {
    const float* z  = (const float*)d_in[0];   // [32768, 1024] f32
    const float* cb = (const float*)d_in[1];   // [256, 1024] f32
    float* out = (float*)d_out;

    float* csq    = (float*)d_ws;                                // 256 f32
    int*   idx_ws = (int*)((char*)d_ws + CODES * sizeof(float)); // 32768 i32

    vq_csq_kernel   <<<CODES,           256, 0, stream>>>(cb, csq);
    vq_argmin_kernel<<<NROWS / ROWS_WG, 256, 0, stream>>>(z, cb, csq, idx_ws);
    vq_gather_kernel<<<NROWS,           256, 0, stream>>>(cb, idx_ws, out,
                                                          out + (size_t)NROWS * DIMS);
}